// GNO_34875134443692
// MI455X (gfx1250) — compile-verified
//
#include <hip/hip_runtime.h>
#include <hip/hip_bf16.h>

#define T_LEN  16384
#define HP_LEN 16448   // reflect-padded signal length (32 + T + 32)
#define NFR    513     // frames (TT)
#define NC2    66      // interleaved spectrum row: 33 real | 33 imag
#define BATCH  32
#define HCH    64
#define CTOT   2048    // BATCH*HCH

typedef __attribute__((ext_vector_type(2))) float v2f;
typedef __attribute__((ext_vector_type(8))) float v8f;

__device__ __forceinline__ v8f wmma_f32(v2f a, v2f b, v8f c) {
  // D = A(16x4) * B(4x16) + C(16x16), fp32, wave32
  return __builtin_amdgcn_wmma_f32_16x16x4_f32(false, a, false, b, (short)0, c,
                                               false, false);
}

// exact period-64 angle reduction: cos/sin(2*pi*p/64)
__device__ __forceinline__ float cos64(int p) {
  return __cosf(0.09817477042f * (float)(p & 63));
}
__device__ __forceinline__ float sin64(int p) {
  return __sinf(0.09817477042f * (float)(p & 63));
}

// ---------------------------------------------------------------- lift (1x1)
__global__ __launch_bounds__(256) void lift_kernel(const float* __restrict__ x,
                                                   const float* __restrict__ lw,
                                                   const float* __restrict__ lb,
                                                   float* __restrict__ hp) {
  const int t = blockIdx.x * 256 + threadIdx.x;
  const int b = blockIdx.y;
  float xv[4];
#pragma unroll
  for (int c = 0; c < 4; ++c) xv[c] = x[((size_t)(b * 4 + c)) * T_LEN + t];
  for (int o = 0; o < HCH; ++o) {
    float acc = lb[o];
#pragma unroll
    for (int c = 0; c < 4; ++c) acc += xv[c] * lw[o * 4 + c];
    hp[((size_t)(b * HCH + o)) * HP_LEN + 32 + t] = acc;
  }
}

// ------------------------------------------------------- refresh reflect pads
// hp[c][p] = x[32-p] = hp[c][64-p];  hp[c][T+32+q] = x[T-2-q] = hp[c][T+30-q]
__global__ __launch_bounds__(64) void pad_kernel(float* __restrict__ hp) {
  const int c = blockIdx.x;
  const int p = threadIdx.x;
  float* hc = hp + (size_t)c * HP_LEN;
  if (p < 32) hc[p] = hc[64 - p];
  else {
    const int q = p - 32;
    hc[T_LEN + 32 + q] = hc[T_LEN + 30 - q];
  }
}

// ---------------------------------------------------------------- proj (1x1)
__global__ __launch_bounds__(256) void proj_kernel(const float* __restrict__ hp,
                                                   const float* __restrict__ pw,
                                                   const float* __restrict__ pb,
                                                   float* __restrict__ out) {
  const int t = blockIdx.x * 256 + threadIdx.x;
  const int b = blockIdx.y;
  float acc[4];
#pragma unroll
  for (int o = 0; o < 4; ++o) acc[o] = pb[o];
  for (int c = 0; c < HCH; ++c) {
    const float hv = hp[((size_t)(b * HCH + c)) * HP_LEN + 32 + t];
#pragma unroll
    for (int o = 0; o < 4; ++o) acc[o] += hv * pw[o * HCH + c];
  }
#pragma unroll
  for (int o = 0; o < 4; ++o) out[((size_t)(b * 4 + o)) * T_LEN + t] = acc[o];
}

// ---------------------------------------------------------------- STFT
// U[c][t][n] (n<33 real, n>=33 imag) = (1/8) sum_j w[j]*hp[t*32+j]*e^{-i2pi f j/64}
// GEMM per wave: A = 16 frames x 64 samples (b64 loads from padded signal),
// B(LDS) = 64 x 80 basis (33 cos | 33 -sin | pad)
__global__ __launch_bounds__(256) void stft_kernel(const float* __restrict__ hp,
                                                   float* __restrict__ U) {
  __shared__ float bs[80 * 64];  // bs[n][j] transposed basis, 20KB
  const int tid = threadIdx.x;
  for (int idx = tid; idx < 80 * 64; idx += 256) {
    const int n = idx >> 6, j = idx & 63;
    const float w = 0.5f - 0.5f * cos64(j);
    float v = 0.0f;
    if (n < 33) v = 0.125f * w * cos64(n * j);
    else if (n < 66) v = -0.125f * w * sin64((n - 33) * j);
    bs[idx] = v;
  }
  __syncthreads();

  const int lane = tid & 31, wid = tid >> 5;
  const int c = blockIdx.x;                 // 0..2047
  const int tile = blockIdx.y * 8 + wid;    // frame tile of 16
  if (tile >= 33) return;
  const int k0 = tile * 16;
  const float* xc = hp + (size_t)c * HP_LEN;

  const int row = lane & 15, half = lane >> 4;
  int frame = k0 + row;
  if (frame > 512) frame = 512;
  const int sbase = frame * 32;             // padded index of sample j=0

  v8f acc[5] = {};
  for (int kk = 0; kk < 16; ++kk) {
    const int kp = kk * 4 + half * 2;       // sample j (even -> 8B aligned)
    const v2f a = *(const v2f*)&xc[sbase + kp];
#pragma unroll
    for (int n = 0; n < 5; ++n) {
      const v2f bb = *(const v2f*)&bs[(n * 16 + row) * 64 + kp];
      acc[n] = wmma_f32(a, bb, acc[n]);
    }
  }
#pragma unroll
  for (int n = 0; n < 5; ++n)
#pragma unroll
    for (int v = 0; v < 8; ++v) {
      const int fr = k0 + v + 8 * half;
      const int col = n * 16 + row;
      if (fr <= 512 && col < NC2)
        U[((size_t)c * NFR + fr) * NC2 + col] = acc[n][v];
    }
}

// ---------------------------------------------------------------- channel mix
// Y[b,o,f,t] = sum_i U[b,i,f,t] * W[o,i,f,t], in place over U ([c][t][66])
// Wave = (t, plane); 8 t's x 2 planes per 512-thread block; block = (f, tblock)
__global__ __launch_bounds__(512) void mix_kernel(const float* __restrict__ W,
                                                  float* __restrict__ U) {
  const int tid = threadIdx.x;
  const int lane = tid & 31, wid = tid >> 5;
  const int f = blockIdx.x;                  // 0..32
  const int tq = wid >> 1, ri = wid & 1;
  const int t = blockIdx.y * 8 + tq;
  if (t >= NFR) return;
  const int po = ri * 33 + f;                // column within interleaved row
  const int row = lane & 15, half = lane >> 4;

  v8f acc[2][4] = {};
  for (int kk = 0; kk < 16; ++kk) {
    const int i = kk * 4 + half * 2;
    v2f a[2];
#pragma unroll
    for (int m = 0; m < 2; ++m) {
      const int b = m * 16 + row;
      const size_t base = (((size_t)(b * HCH + i)) * NFR + t) * NC2 + po;
      a[m].x = U[base];
      a[m].y = U[base + (size_t)NFR * NC2];
    }
    v2f bb[4];
#pragma unroll
    for (int n = 0; n < 4; ++n) {
      const int o = n * 16 + row;
      const size_t wb = (((size_t)(o * HCH + i)) * 33 + f) * NFR + t;
      bb[n].x = W[wb];
      bb[n].y = W[wb + (size_t)33 * NFR];
    }
#pragma unroll
    for (int m = 0; m < 2; ++m)
#pragma unroll
      for (int n = 0; n < 4; ++n) acc[m][n] = wmma_f32(a[m], bb[n], acc[m][n]);
  }
#pragma unroll
  for (int m = 0; m < 2; ++m)
#pragma unroll
    for (int n = 0; n < 4; ++n)
#pragma unroll
      for (int v = 0; v < 8; ++v) {
        const int b = m * 16 + v + 8 * half;
        const int o = n * 16 + row;
        U[(((size_t)(b * HCH + o)) * NFR + t) * NC2 + po] = acc[m][n][v];
      }
}

// ---------------------------------------------------------------- ISTFT + add
// frames[k][j] = (1/8)*w[j]*sum_f alpha_f(cos*Yr - sin*Yi); overlap-add /wsq
// Wave = one channel c, 16 frames [kb..kb+15]; emits t for k in [kb+1..kb+15]
__global__ __launch_bounds__(256) void istft_kernel(const float* __restrict__ Y,
                                                    float* __restrict__ hp) {
  __shared__ float bi[64 * 68];          // bi[j][k] transposed inverse basis
  __shared__ float ov[8 * 16 * 64];      // per-wave frame buffer, 32KB
  const int tid = threadIdx.x;
  for (int idx = tid; idx < 64 * 68; idx += 256) {
    const int j = idx / 68, k = idx % 68;
    const float w = 0.5f - 0.5f * cos64(j);
    float v = 0.0f;
    if (k < 33) {
      const float al = (k == 0 || k == 32) ? 1.0f : 2.0f;
      v = 0.125f * al * w * cos64(k * j);
    } else if (k < 66) {
      const int fq = k - 33;
      const float al = (fq == 0 || fq == 32) ? 1.0f : 2.0f;
      v = -0.125f * al * w * sin64(fq * j);
    }
    bi[idx] = v;
  }
  __syncthreads();

  const int lane = tid & 31, wid = tid >> 5;
  const int c = blockIdx.y * 8 + wid;      // 0..2047
  const int kb = 15 * blockIdx.x;          // frame tile base (overlap by 1)
  const float* yc = Y + (size_t)c * NFR * NC2;
  const int row = lane & 15, half = lane >> 4;
  int frame = kb + row;
  if (frame > 512) frame = 512;
  const size_t fb = (size_t)frame * NC2;

  v8f acc[4] = {};
  for (int kk = 0; kk < 16; ++kk) {        // K = 0..63: unconditional b64 loads
    const int kp = kk * 4 + half * 2;
    const v2f a = *(const v2f*)&yc[fb + kp];
#pragma unroll
    for (int n = 0; n < 4; ++n) {
      const v2f bb = *(const v2f*)&bi[(n * 16 + row) * 68 + kp];
      acc[n] = wmma_f32(a, bb, acc[n]);
    }
  }
  {                                        // K = 64..67: tail (cols 66,67 pad)
    const int kp = 64 + half * 2;
    v2f a = {0.0f, 0.0f};
    if (kp < NC2) a = *(const v2f*)&yc[fb + kp];
#pragma unroll
    for (int n = 0; n < 4; ++n) {
      const v2f bb = *(const v2f*)&bi[(n * 16 + row) * 68 + kp];
      acc[n] = wmma_f32(a, bb, acc[n]);
    }
  }
  // scatter D tiles (frame, j) into this wave's LDS frame buffer
  float* myov = &ov[wid * 1024];
#pragma unroll
  for (int n = 0; n < 4; ++n)
#pragma unroll
    for (int v = 0; v < 8; ++v) {
      const int fr = v + 8 * half;  // 0..15 local frame
      myov[fr * 64 + n * 16 + row] = acc[n][v];
    }
  __syncthreads();

  // overlap-add: lane = j0 in [0,32)
  const int j0 = lane;
  const float w0 = 0.5f - 0.5f * cos64(j0);
  const float w1 = 0.5f - 0.5f * cos64(j0 + 32);
  const float invw = 1.0f / (w0 * w0 + w1 * w1);
  float* hc = hp + (size_t)c * HP_LEN + 32;
  for (int m = 1; m < 16; ++m) {
    const int k = kb + m;
    if (k > 512) break;
    const int t = 32 * (k - 1) + j0;
    const float val = myov[m * 64 + j0] + myov[(m - 1) * 64 + j0 + 32];
    hc[t] += val * invw;
  }
}

extern "C" void kernel_launch(void* const* d_in, const int* in_sizes, int n_in,
                              void* d_out, int out_size, void* d_ws, size_t ws_size,
                              hipStream_t stream) {
  const float* x  = (const float*)d_in[0];
  const float* lw = (const float*)d_in[1];
  const float* lb = (const float*)d_in[2];
  const float* g1 = (const float*)d_in[3];
  const float* g2 = (const float*)d_in[4];
  const float* pw = (const float*)d_in[5];
  const float* pb = (const float*)d_in[6];
  float* out = (float*)d_out;

  float* hb = (float*)d_ws;                 // 2048*16448 floats (padded h)
  float* Ub = hb + (size_t)CTOT * HP_LEN;   // 2048*513*66 floats (+ tail slack)

  lift_kernel<<<dim3(T_LEN / 256, BATCH), 256, 0, stream>>>(x, lw, lb, hb);
  pad_kernel<<<dim3(CTOT), 64, 0, stream>>>(hb);
  for (int layer = 0; layer < 2; ++layer) {
    const float* W = layer ? g2 : g1;
    stft_kernel<<<dim3(CTOT, 5), 256, 0, stream>>>(hb, Ub);
    mix_kernel<<<dim3(33, 65), 512, 0, stream>>>(W, Ub);
    istft_kernel<<<dim3(35, CTOT / 8), 256, 0, stream>>>(Ub, hb);
    if (layer == 0) pad_kernel<<<dim3(CTOT), 64, 0, stream>>>(hb);
  }
  proj_kernel<<<dim3(T_LEN / 256, BATCH), 256, 0, stream>>>(hb, pw, pb, out);
}